// AttentionEncoder_5978594476334
// MI455X (gfx1250) — compile-verified
//
#include <hip/hip_runtime.h>
#include <hip/hip_bf16.h>
#include <math.h>

// Problem constants (match reference)
#define BB 256
#define LL 64
#define DD 256
#define HH 512

typedef float v2f __attribute__((ext_vector_type(2)));
typedef float v8f __attribute__((ext_vector_type(8)));
typedef unsigned int v4u __attribute__((ext_vector_type(4)));
typedef int v4i __attribute__((ext_vector_type(4)));
typedef int v8i __attribute__((ext_vector_type(8)));

__device__ __forceinline__ float sigmoidf_(float v) {
    return 1.0f / (1.0f + __expf(-v));
}

// ---------------------------------------------------------------------------
// TDM: DMA a 32-row x 64-col fp32 tile (row stride = row_stride_elems) from
// global memory into LDS at lds_byte_addr. D# bit layout per CDNA5 ISA ch.8:
//   group0: [1:0]=count=1 | [63:32]=lds_addr | [120:64]=global_addr |
//           [127:126]=type=2
//   group1: [17:16]=data_size=2 (4B) | tensor_dim0/1 = 0x10000 (never clips;
//           all our tiles are in-bounds) | tile_dim0=64 | tile_dim1=32 |
//           tensor_dim0_stride = row stride
//   groups 2/3 + extra group: zero (2-D tile; tile_dim2=0, no iterate/
//   gather/multicast)
// ---------------------------------------------------------------------------
__device__ __forceinline__ void tdm_load_tile_32x64(const float* src,
                                                    unsigned row_stride_elems,
                                                    unsigned lds_byte_addr) {
    const unsigned long long ga = (unsigned long long)(uintptr_t)src;
    v4u g0;
    g0[0] = 1u;                                   // count=1, user descriptor
    g0[1] = lds_byte_addr;                        // LDS destination
    g0[2] = (unsigned)ga;                         // global_addr[31:0]
    g0[3] = (unsigned)(ga >> 32) | (2u << 30);    // global_addr[56:32]|type=2
    v8i g1;
    g1[0] = 0x20000;                 // workgroup_mask=0, data_size=2 (4 bytes)
    g1[1] = 0;                       // tensor_dim0 = 0x10000 -> low16 = 0
    g1[2] = 1;                       // dim0 hi16 = 1 | tensor_dim1 low16 = 0
    g1[3] = 1 | (64 << 16);          // tensor_dim1 hi16 = 1 | tile_dim0 = 64
    g1[4] = 32;                      // tile_dim1 = 32, tile_dim2 = 0
    g1[5] = (int)row_stride_elems;   // tensor_dim0_stride (low 32 of 48)
    g1[6] = 0;                       // stride hi | tensor_dim1_stride lo = 0
    g1[7] = 0;
    const v4i z4 = {0, 0, 0, 0};
    const v8i z8 = {0, 0, 0, 0, 0, 0, 0, 0};
    // amdgpu-toolchain clang-23 signature (6 args):
    //   (v4u g0, v8i g1, v4i g2, v4i g3, v8i extra, i32 cpol)
    __builtin_amdgcn_tensor_load_to_lds(g0, g1, z4, z4, z8, 0);
}

// ---------------------------------------------------------------------------
// Kernel A: a[b,:] = softmax_D( sum_l x[b,l,:]*Wx[l] ).
// The recurrent additive term (h.Wh + c.Wc + ba) is constant along the softmax
// axis -> softmax is shift-invariant -> attention weights are timestep-
// independent. Emit att[b,t,:] = a[b,:] for all t, cache a, and precompute
// wi[b,l,:] = a[b,:] * x[b,l,:] for ALL timesteps (valid since a is constant
// in t) so the step kernel's input-GEMM operand is a pure strided copy.
// ---------------------------------------------------------------------------
__global__ void __launch_bounds__(DD)
attn_softmax_kernel(const float* __restrict__ x,    // (B,L,D)
                    const float* __restrict__ Wa,   // (2H+L,)
                    float* __restrict__ att_out,    // (B,L,D)
                    float* __restrict__ a_ws,       // (B,D)
                    float* __restrict__ wi_ws) {    // (B,L,D)
    const int b = blockIdx.x;
    const int d = threadIdx.x;

    __shared__ float sWx[LL];
    __shared__ float red[DD];

    if (d < LL) sWx[d] = Wa[2 * HH + d];   // Wx = Wa[2H:]
    __syncthreads();

    const float* xb = x + (size_t)b * LL * DD;
    float ex = 0.0f;
#pragma unroll 8
    for (int l = 0; l < LL; ++l) ex += xb[l * DD + d] * sWx[l];

    red[d] = ex;
    __syncthreads();
    for (int s = DD / 2; s > 0; s >>= 1) {
        if (d < s) red[d] = fmaxf(red[d], red[d + s]);
        __syncthreads();
    }
    const float mx = red[0];
    __syncthreads();

    const float e = __expf(ex - mx);
    red[d] = e;
    __syncthreads();
    for (int s = DD / 2; s > 0; s >>= 1) {
        if (d < s) red[d] += red[d + s];
        __syncthreads();
    }
    const float a = e / red[0];

    a_ws[b * DD + d] = a;
    float* ob = att_out + (size_t)b * LL * DD + d;
    float* wb = wi_ws + (size_t)b * LL * DD + d;
#pragma unroll 4
    for (int t = 0; t < LL; ++t) {
        ob[t * DD] = a;                    // att output (broadcast over t)
        wb[t * DD] = a * xb[t * DD + d];   // weighted input, all timesteps
    }
}

// ---------------------------------------------------------------------------
// Kernel B: one LSTM timestep (launched 64x = the true recurrence).
//   gates = h_prev @ W_hh^T (K=512) + wi_t @ W_ih^T (K=256) + bias
// Block = 128 threads = 4 waves; block tile = 32 batch rows x 64 h-columns.
// Each wave: 16 h-columns x 4 gates x 2 row-subtiles = 8 v8f accumulators,
// so every B fragment (global b64) feeds TWO v_wmma_f32_16x16x4_f32 -- halves
// the VMEM pressure per matrix op vs a 16-row tile.
// A tiles (32x64 chunks of h_prev / wi_t) are DMA'd into double-buffered LDS
// by the Tensor Data Mover: wave 0 prefetches chunk q+1 while all waves
// consume chunk q, then s_wait_tensorcnt 0 + barrier.
// grid = 8 (M) x 8 (col) = 64 blocks.
// ---------------------------------------------------------------------------
__global__ void __launch_bounds__(128)
lstm_step_kernel(const float* __restrict__ wi,      // (B,L,D) weighted input
                 const float* __restrict__ W_ih,    // (4H,D) row-major
                 const float* __restrict__ W_hh,    // (4H,H) row-major
                 const float* __restrict__ b_ih,    // (4H,)
                 const float* __restrict__ b_hh,    // (4H,)
                 const float* __restrict__ h_prev,  // (B,H)
                 const float* __restrict__ c_prev,  // (B,H)
                 float* __restrict__ h_next,        // (B,H)
                 float* __restrict__ c_next,        // (B,H)
                 float* __restrict__ enc_out,       // (B,L,H)
                 int t) {
    const int mtile = blockIdx.x & 7;    // 8 tiles of 32 batch rows
    const int ntile = blockIdx.x >> 3;   // 8 tiles of 64 h-columns
    const int bm0   = mtile * 32;

    const int tid  = threadIdx.x;
    const int wave = tid >> 5;
    const int lane = tid & 31;
    const int n0    = ntile * 64 + wave * 16;  // this wave's h-column base
    const int nlane = lane & 15;               // N index within tile
    const int mrow  = lane & 15;               // A-matrix M index (per ISA)
    const int koff  = (lane >> 4) * 2;         // A/B K sub-offset (per ISA)

    __shared__ float shA[2][32 * 64];          // double-buffered 32x64 chunk
    const unsigned lds0 = (unsigned)(uintptr_t)&shA[0][0];
    const unsigned lds1 = (unsigned)(uintptr_t)&shA[1][0];

    // 8 accumulators: [row-subtile][gate], seeded with bias (column-only).
    v8f acc[2][4];
#pragma unroll
    for (int g = 0; g < 4; ++g) {
        const int j = g * HH + n0 + nlane;
        const float bias = b_ih[j] + b_hh[j];
        v8f c;
#pragma unroll
        for (int s = 0; s < 8; ++s) c[s] = bias;
        acc[0][g] = c;
        acc[1][g] = c;
    }

    // 12 K-chunks of 64: q = 0..7 -> h_prev/W_hh (K=512),
    //                    q = 8..11 -> wi_t/W_ih (K=256).
    const float* const hbase  = h_prev + (size_t)bm0 * HH;
    const float* const wibase = wi + ((size_t)bm0 * LL + t) * DD;

    // Prologue: TDM chunk 0 into buffer 0.
    if (wave == 0) {
        tdm_load_tile_32x64(hbase, HH, lds0);
        __builtin_amdgcn_s_wait_tensorcnt(0);
    }
    __syncthreads();

    int cur = 0;
    for (int q = 0; q < 12; ++q) {
        // Prefetch next chunk into the other buffer while we compute.
        if (wave == 0 && q + 1 < 12) {
            const int qn = q + 1;
            const float* src = (qn < 8) ? (hbase + qn * 64)
                                        : (wibase + (qn - 8) * 64);
            const unsigned stride = (qn < 8) ? HH : (LL * DD);
            tdm_load_tile_32x64(src, stride, (cur ^ 1) ? lds1 : lds0);
        }

        // Weight operand for this chunk.
        const float* Wsel = (q < 8) ? W_hh : W_ih;
        const unsigned wstride = (q < 8) ? HH : DD;
        const int kc = (q < 8) ? q * 64 : (q - 8) * 64;

        const float* wg0 = Wsel + (size_t)(0 * HH + n0 + nlane) * wstride + kc + koff;
        const float* wg1 = Wsel + (size_t)(1 * HH + n0 + nlane) * wstride + kc + koff;
        const float* wg2 = Wsel + (size_t)(2 * HH + n0 + nlane) * wstride + kc + koff;
        const float* wg3 = Wsel + (size_t)(3 * HH + n0 + nlane) * wstride + kc + koff;
        const float* shc = &shA[cur][0];

#pragma unroll 4
        for (int kk = 0; kk < 64; kk += 4) {
            v2f a0, a1;
            a0.x = shc[mrow * 64 + kk + koff];
            a0.y = shc[mrow * 64 + kk + koff + 1];
            a1.x = shc[(16 + mrow) * 64 + kk + koff];
            a1.y = shc[(16 + mrow) * 64 + kk + koff + 1];

            v2f b0, b1, b2, b3;
            b0.x = wg0[kk]; b0.y = wg0[kk + 1];
            b1.x = wg1[kk]; b1.y = wg1[kk + 1];
            b2.x = wg2[kk]; b2.y = wg2[kk + 1];
            b3.x = wg3[kk]; b3.y = wg3[kk + 1];

            acc[0][0] = __builtin_amdgcn_wmma_f32_16x16x4_f32(
                false, a0, false, b0, (short)0, acc[0][0], false, false);
            acc[1][0] = __builtin_amdgcn_wmma_f32_16x16x4_f32(
                false, a1, false, b0, (short)0, acc[1][0], false, false);
            acc[0][1] = __builtin_amdgcn_wmma_f32_16x16x4_f32(
                false, a0, false, b1, (short)0, acc[0][1], false, false);
            acc[1][1] = __builtin_amdgcn_wmma_f32_16x16x4_f32(
                false, a1, false, b1, (short)0, acc[1][1], false, false);
            acc[0][2] = __builtin_amdgcn_wmma_f32_16x16x4_f32(
                false, a0, false, b2, (short)0, acc[0][2], false, false);
            acc[1][2] = __builtin_amdgcn_wmma_f32_16x16x4_f32(
                false, a1, false, b2, (short)0, acc[1][2], false, false);
            acc[0][3] = __builtin_amdgcn_wmma_f32_16x16x4_f32(
                false, a0, false, b3, (short)0, acc[0][3], false, false);
            acc[1][3] = __builtin_amdgcn_wmma_f32_16x16x4_f32(
                false, a1, false, b3, (short)0, acc[1][3], false, false);
        }

        // Ensure the prefetched tile has landed, then publish to all waves.
        if (wave == 0) __builtin_amdgcn_s_wait_tensorcnt(0);
        __syncthreads();
        cur ^= 1;
    }

    // ---- Elementwise LSTM update + stores (C layout: M = s + 8*(lane>=16)) -
    const int rbase = (lane >> 4) * 8;
#pragma unroll
    for (int ms = 0; ms < 2; ++ms) {
#pragma unroll
        for (int s = 0; s < 8; ++s) {
            const int row = bm0 + ms * 16 + rbase + s;
            const int col = n0 + nlane;
            const float iv = acc[ms][0][s];
            const float fv = acc[ms][1][s];
            const float gv = acc[ms][2][s];
            const float ov = acc[ms][3][s];
            const float cp = c_prev[row * HH + col];
            const float cn = sigmoidf_(fv) * cp + sigmoidf_(iv) * tanhf(gv);
            const float hn = sigmoidf_(ov) * tanhf(cn);
            c_next[row * HH + col] = cn;
            h_next[row * HH + col] = hn;
            enc_out[((size_t)row * LL + t) * HH + col] = hn;
        }
    }
}

// ---------------------------------------------------------------------------
extern "C" void kernel_launch(void* const* d_in, const int* in_sizes, int n_in,
                              void* d_out, int out_size, void* d_ws, size_t ws_size,
                              hipStream_t stream) {
    const float* x    = (const float*)d_in[0];   // (B,L,D)
    const float* h0   = (const float*)d_in[1];   // (B,H)
    const float* c0   = (const float*)d_in[2];   // (B,H)
    const float* W_ih = (const float*)d_in[3];   // (4H,D)
    const float* W_hh = (const float*)d_in[4];   // (4H,H)
    const float* b_ih = (const float*)d_in[5];   // (4H,)
    const float* b_hh = (const float*)d_in[6];   // (4H,)
    const float* Wa   = (const float*)d_in[7];   // (2H+L,)
    // d_in[8] = ba: softmax shift-invariance makes it (and Wh/Wc) a no-op.

    float* out = (float*)d_out;
    float* att = out;                                // (B,L,D)
    float* enc = out + (size_t)BB * LL * DD;         // (B,L,H)

    // Workspace (floats): a (B*D) | wi (B*L*D) | h ping-pong | c ping-pong
    // = 65536 + 4194304 + 4*131072 floats ~= 19.2 MB.
    float* ws    = (float*)d_ws;
    float* a_ws  = ws;
    float* wi_ws = a_ws + BB * DD;
    float* hb0   = wi_ws + (size_t)BB * LL * DD;
    float* hb1   = hb0 + BB * HH;
    float* cb0   = hb1 + BB * HH;
    float* cb1   = cb0 + BB * HH;
    float* hbuf[2] = {hb0, hb1};
    float* cbuf[2] = {cb0, cb1};

    // Seed the recurrence state (graph-capture-safe d2d async copies).
    (void)hipMemcpyAsync(hbuf[0], h0, sizeof(float) * BB * HH,
                         hipMemcpyDeviceToDevice, stream);
    (void)hipMemcpyAsync(cbuf[0], c0, sizeof(float) * BB * HH,
                         hipMemcpyDeviceToDevice, stream);

    // Attention weights (timestep-invariant), att output, weighted inputs.
    attn_softmax_kernel<<<BB, DD, 0, stream>>>(x, Wa, att, a_ws, wi_ws);

    // Sequential LSTM scan: 64 dependent launches, ping-ponging h/c.
    for (int t = 0; t < LL; ++t) {
        lstm_step_kernel<<<64, 128, 0, stream>>>(
            wi_ws, W_ih, W_hh, b_ih, b_hh,
            hbuf[t & 1], cbuf[t & 1], hbuf[(t + 1) & 1], cbuf[(t + 1) & 1],
            enc, t);
    }
}